// StructNConv2D_d_with_s_14697378087431
// MI455X (gfx1250) — compile-verified
//
#include <hip/hip_runtime.h>
#include <hip/hip_bf16.h>

// Fused StructNConv2D_d_with_s for MI455X (gfx1250, wave32).
// Stage 1 (spatial 3x3 weighted reduce) + Stage 2 (16x16 channel matmul via
// V_WMMA_F32_16X16X4_F32 chained over K) in a single memory pass.
// Memory-bound: ~210 MB minimum traffic -> ~9 us at 23.3 TB/s.

#define EPSF 1e-20f
#define BB 4
#define CC 16
#define HH 256
#define WW 256
#define KT 9   // 3x3 taps

typedef __attribute__((ext_vector_type(2))) float v2f;
typedef __attribute__((ext_vector_type(8))) float v8f;

__global__ __launch_bounds__(256)
void structnconv_fused(const float* __restrict__ dmap,
                       const float* __restrict__ cdmap,
                       const float* __restrict__ sprod,   // (B,C,9,H,W)
                       const float* __restrict__ swv,     // (C,9) flattened spatial_weight
                       const float* __restrict__ cwv,     // (O=16, C=16)
                       float* __restrict__ dout,
                       float* __restrict__ cdout)
{
    __shared__ float sw_lds[CC * KT];     // sw[c][k]
    __shared__ float wt_lds[CC * CC];     // Wt[c][o] = cw[o][c]  (B-matrix, row = K = c)
    __shared__ float Xp[8][16][17];       // per-wave [pixel][channel], padded (64-bank safe)
    __shared__ float Xc[8][16][17];

    const int tid  = threadIdx.x;
    const int lane = tid & 31;
    const int wv   = tid >> 5;
    const int half = lane >> 4;           // 0: lanes 0-15, 1: lanes 16-31
    const int ln   = lane & 15;

    if (tid < CC * KT) sw_lds[tid] = swv[tid];
    if (tid < CC * CC) wt_lds[tid] = cwv[(tid & 15) * CC + (tid >> 4)]; // transpose
    __syncthreads();

    // Global normalizers (tiny LDS scans, once per block).
    float Ssw = 0.f;
    for (int i = 0; i < CC * KT; ++i) Ssw += sw_lds[i];
    float Scw = 0.f;
    for (int i = 0; i < CC * CC; ++i) Scw += wt_lds[i];
    const float invSsw = 1.f / (Ssw + EPSF);
    const float invScw = 1.f / (Scw + EPSF);

    // B fragments for V_WMMA_F32_16X16X4_F32 (4x16 per step):
    // VGPR0: lanes 0-15 row K=4k+0, lanes 16-31 row K=4k+2
    // VGPR1: lanes 0-15 row K=4k+1, lanes 16-31 row K=4k+3
    v2f bfrag[4];
#pragma unroll
    for (int k = 0; k < 4; ++k) {
        int r0 = 4 * k + 2 * half;
        bfrag[k][0] = wt_lds[(r0 + 0) * CC + ln];
        bfrag[k][1] = wt_lds[(r0 + 1) * CC + ln];
    }

    // One wave per 16-pixel tile: 4*256*16 = 16384 tiles = 2048 blocks * 8 waves.
    const int gw = blockIdx.x * 8 + wv;
    const int w0 = (gw & 15) * 16;
    const int h  = (gw >> 4) & 255;
    const int b  = gw >> 12;

    // ---------------- Stage 1: spatial 3x3 weighted reduce -----------------
    // Lane covers 8 (pixel,channel) pairs; half-wave strides give coalesced
    // 64B global transactions along w.
#pragma unroll
    for (int i = 0; i < 8; ++i) {
        const int idx = lane + 32 * i;       // 0..255
        const int c   = idx >> 4;
        const int p   = idx & 15;
        const int w   = w0 + p;

        const float* dbase  = dmap  + ((b * CC + c) * HH) * WW;
        const float* cdbase = cdmap + ((b * CC + c) * HH) * WW;
        const float* spbase = sprod + ((size_t)(b * CC + c) * KT) * HH * WW
                                    + (size_t)h * WW + w;
        float den = 0.f, nom = 0.f;
#pragma unroll
        for (int di = -1; di <= 1; ++di) {
            const int hh = h + di;
            const bool hok = (unsigned)hh < (unsigned)HH;
#pragma unroll
            for (int dj = -1; dj <= 1; ++dj) {
                const int ww = w + dj;
                const bool ok = hok && ((unsigned)ww < (unsigned)WW);
                const int k  = (di + 1) * 3 + (dj + 1);
                const float cdv = ok ? cdbase[hh * WW + ww] : 0.f; // zero-pad
                const float dv  = ok ? dbase [hh * WW + ww] : 0.f;
                const float sp  = spbase[(size_t)k * HH * WW];     // center-indexed
                const float t   = cdv * sp * sw_lds[c * KT + k];
                den += t;
                nom += t * dv;
            }
        }
        const float dsp = nom / (den + EPSF);       // d_spatial
        const float csp = den * invSsw;             // cd_spatial
        Xp[wv][p][c] = csp * dsp;                   // A for nom2
        Xc[wv][p][c] = csp;                         // A for denom2
    }
    // Wave-private LDS: DS ops are in-order per wave, no barrier needed.

    // ---------------- Stage 2: 16x16 channel matmul on the matrix pipe -----
    v8f accN = {0.f, 0.f, 0.f, 0.f, 0.f, 0.f, 0.f, 0.f};
    v8f accD = {0.f, 0.f, 0.f, 0.f, 0.f, 0.f, 0.f, 0.f};
#if defined(__gfx1250__) && __has_builtin(__builtin_amdgcn_wmma_f32_16x16x4_f32)
#pragma unroll
    for (int k = 0; k < 4; ++k) {
        const int c0 = 4 * k + 2 * half;            // A layout: 16x4 f32
        v2f ap, ac;
        ap[0] = Xp[wv][ln][c0];  ap[1] = Xp[wv][ln][c0 + 1];
        ac[0] = Xc[wv][ln][c0];  ac[1] = Xc[wv][ln][c0 + 1];
        accN = __builtin_amdgcn_wmma_f32_16x16x4_f32(
                   false, ap, false, bfrag[k], (short)0, accN, false, false);
        accD = __builtin_amdgcn_wmma_f32_16x16x4_f32(
                   false, ac, false, bfrag[k], (short)0, accD, false, false);
    }
#else
    // VALU fallback (host pass / toolchains without the f32 WMMA builtin).
#pragma unroll
    for (int j = 0; j < 8; ++j) {
        const int pj = j + half * 8;
        float n2 = 0.f, d2 = 0.f;
        for (int c = 0; c < CC; ++c) {
            const float wgt = wt_lds[c * CC + ln];
            n2 += Xp[wv][pj][c] * wgt;
            d2 += Xc[wv][pj][c] * wgt;
        }
        accN[j] = n2;
        accD[j] = d2;
    }
#endif

    // D layout: VGPR j = row M = j + 8*half (pixel), column N = ln (out ch).
    // Transpose through LDS so the final stores are coalesced along w.
#pragma unroll
    for (int j = 0; j < 8; ++j) {
        const int pj   = j + half * 8;
        const float dn = accD[j];
        Xp[wv][pj][ln] = accN[j] / (dn + EPSF);     // d_out
        Xc[wv][pj][ln] = dn * invScw;               // cd_out (devalue_conf = 1)
    }
#pragma unroll
    for (int i = 0; i < 8; ++i) {
        const int idx = lane + 32 * i;
        const int o   = idx >> 4;
        const int p   = idx & 15;
        const int off = ((b * CC + o) * HH + h) * WW + w0 + p;
        dout[off]  = Xp[wv][p][o];
        cdout[off] = Xc[wv][p][o];
    }
}

extern "C" void kernel_launch(void* const* d_in, const int* in_sizes, int n_in,
                              void* d_out, int out_size, void* d_ws, size_t ws_size,
                              hipStream_t stream) {
    (void)in_sizes; (void)n_in; (void)d_ws; (void)ws_size; (void)out_size;
    const float* dmap  = (const float*)d_in[0];   // d
    const float* cdmap = (const float*)d_in[1];   // cd
    // d_in[2] = s, d_in[3] = cs: unused by the reference -> never loaded.
    const float* sprod = (const float*)d_in[4];   // s_prod_roll
    const float* swv   = (const float*)d_in[5];   // spatial_weight (C,1,9,1,1)
    const float* cwv   = (const float*)d_in[6];   // channel_weight (C,C,1,1)

    float* dout  = (float*)d_out;
    float* cdout = dout + (size_t)BB * CC * HH * WW;

    const int tiles  = BB * HH * (WW / 16);       // 16384
    const int blocks = tiles / 8;                 // 2048 (8 waves / block)
    structnconv_fused<<<blocks, 256, 0, stream>>>(dmap, cdmap, sprod, swv, cwv,
                                                  dout, cdout);
}